// MidiExtractor_72576357367954
// MI455X (gfx1250) — compile-verified
//
#include <hip/hip_runtime.h>
#include <hip/hip_bf16.h>
#include <math.h>

typedef __attribute__((ext_vector_type(2))) float v2f;
typedef __attribute__((ext_vector_type(8))) float v8f;

#define WMMA_F32X4(a, b, c) \
  __builtin_amdgcn_wmma_f32_16x16x4_f32(false, (a), false, (b), (short)0, (c), false, false)

constexpr int  B_   = 16;
constexpr int  T_   = 6000;
constexpr int  H_   = 256;
constexpr int  N_   = T_ + 1;   // 6001
constexpr int  NCLS = 104;
constexpr int  NCP  = 128;      // pitch cols padded for WMMA
constexpr float THR = 0.5f;
constexpr int  MIN_GAP_ = 9;

// ---------------------------------------------------------------------------
// 0) Pack a [K][C] weight matrix into WMMA-B-friendly layout:
//    out[(k/2)*Cp + col] = {w[k][col], w[k+1][col]}  (zero-padded col >= C)
//    One lane's B fragment then = single global_load_b64, coalesced across lanes.
// ---------------------------------------------------------------------------
__global__ void k_packw(const float* __restrict__ w, float* __restrict__ out,
                        int K, int C, int Cp) {
  long long idx = (long long)blockIdx.x * blockDim.x + threadIdx.x;
  long long total = (long long)(K >> 1) * Cp;
  if (idx >= total) return;
  int col = (int)(idx % Cp);
  int kp = (int)(idx / Cp);
  float x0 = (col < C) ? w[(long long)(2 * kp) * C + col] : 0.f;
  float x1 = (col < C) ? w[(long long)(2 * kp + 1) * C + col] : 0.f;
  v2f v; v.x = x0; v.y = x1;
  ((v2f*)out)[idx] = v;
}

// ---------------------------------------------------------------------------
// 1) Fused GEMV: boundary logits (clip +/-16), sigmoid probs, attn gate
// ---------------------------------------------------------------------------
__global__ __launch_bounds__(256) void k_gemv(
    const float* __restrict__ feat,
    const float* __restrict__ w_bd, const float* __restrict__ b_bd,
    const float* __restrict__ w_attn, const float* __restrict__ b_attn,
    float* __restrict__ bd_logits, float* __restrict__ probs,
    float* __restrict__ attn_mean) {
  int wave = threadIdx.x >> 5;
  int lane = threadIdx.x & 31;
  long long row = (long long)blockIdx.x * 8 + wave;
  if (row >= (long long)B_ * T_) return;
  const float* f = feat + row * H_;
  float s0 = 0.f, s1 = 0.f;
  #pragma unroll
  for (int i = lane; i < H_; i += 32) {
    float v = f[i];
    s0 = fmaf(v, w_bd[i], s0);
    s1 = fmaf(v, w_attn[i], s1);
  }
  #pragma unroll
  for (int off = 16; off > 0; off >>= 1) {
    s0 += __shfl_down(s0, off, 32);
    s1 += __shfl_down(s1, off, 32);
  }
  if (lane == 0) {
    float lg = s0 + b_bd[0];
    lg = fminf(16.f, fmaxf(-16.f, lg));
    bd_logits[row] = lg;
    probs[row] = 1.f / (1.f + __expf(-lg));
    float am = s1 + b_attn[0];
    attn_mean[row] = 1.f / (1.f + __expf(-am));
  }
}

// ---------------------------------------------------------------------------
// 2) Serial regulate_boundary + cumsum, one lane per batch row (16 rows)
// ---------------------------------------------------------------------------
__global__ void k_boundary(const float* __restrict__ probs,
                           const int* __restrict__ non_padding,
                           int* __restrict__ res, int* __restrict__ mel2note,
                           float* __restrict__ out_bd, float* __restrict__ out_len) {
  int b = threadIdx.x;
  if (b >= B_) return;
  const float* p = probs + (long long)b * T_;
  int* r = res + (long long)b * T_;
  for (int j = 0; j < T_; ++j) r[j] = 0;
  int start = -1, last = -1, best_i = 0;
  float best_v = -1e30f;
  for (int j = 0; j < T_; ++j) {
    float pj = p[j];
    bool is_one = pj > THR;
    bool opening = is_one && (start < 0);
    bool upd = opening || (is_one && start >= 0 && pj > best_v);
    if (upd) { best_i = j; best_v = pj; }
    if (opening) start = j;
    bool close = (!is_one) && (start >= 0);
    if (close) {
      bool merge = ((best_i - last) < MIN_GAP_) && (last > 0);
      int mid = (int)rintf((float)(best_i + last) * 0.5f);  // half-to-even
      int fi = merge ? mid : best_i;
      if (merge) r[last] = 0;
      r[fi] = 1;
      last = fi;
      start = -1;
    }
  }
  r[0] = 0;
  int lens = 0;
  const int* np = non_padding + (long long)b * T_;
  for (int j = 0; j < T_; ++j) lens += np[j];
  for (int j = lens - 1; j < T_; ++j)
    if (j >= 0) r[j] = 0;
  int c = 0;
  int* m2n = mel2note + (long long)b * T_;
  float* ob = out_bd + (long long)b * T_;
  for (int j = 0; j < T_; ++j) {
    c += r[j];
    m2n[j] = c;
    ob[j] = (float)r[j];
  }
  out_len[b] = (float)(c + 1);
}

// ---------------------------------------------------------------------------
// 3) Zero / scatter (segment sum) / normalize
// ---------------------------------------------------------------------------
__global__ void k_zero(float* __restrict__ p, long long n) {
  long long i = (long long)blockIdx.x * blockDim.x + threadIdx.x;
  if (i < n) p[i] = 0.f;
}

__global__ void k_scatter(const float* __restrict__ feat,
                          const float* __restrict__ attn_mean,
                          const int* __restrict__ mel2note,
                          float* __restrict__ agg, float* __restrict__ denom) {
  long long idx = (long long)blockIdx.x * blockDim.x + threadIdx.x;  // over B*T*H
  if (idx >= (long long)B_ * T_ * H_) return;
  int h = (int)(idx & (H_ - 1));
  long long bt = idx >> 8;
  int b = (int)(bt / T_);
  float am = attn_mean[bt];
  int seg = mel2note[bt];
  atomicAdd(&agg[((long long)b * N_ + seg) * H_ + h], feat[idx] * am);
  if (h == 0) atomicAdd(&denom[(long long)b * N_ + seg], am);
}

__global__ void k_div(float* __restrict__ agg, const float* __restrict__ denom) {
  long long idx = (long long)blockIdx.x * blockDim.x + threadIdx.x;  // over B*N*H
  if (idx >= (long long)B_ * N_ * H_) return;
  agg[idx] = agg[idx] / (denom[idx >> 8] + 1e-5f);
}

// ---------------------------------------------------------------------------
// 4) K=3 conv (256->256) as WMMA GEMM over 16-position tiles.
//    Weights pre-packed: Wp[(tap*128 + k/2)*256 + col] = {W[tap][k][col], W[tap][k+1][col]}
//    MODE 0: out = x + leaky(LayerNorm(conv(x)+b))      (residual block)
//    MODE 1: out = conv(x)+b                            (post conv)
// ---------------------------------------------------------------------------
template <int MODE>
__global__ __launch_bounds__(256) void k_conv(
    const float* __restrict__ x, const float* __restrict__ Wp,
    const float* __restrict__ bias, const float* __restrict__ lng,
    const float* __restrict__ lnb, float* __restrict__ out) {
  __shared__ float lin[18 * H_];                       // rows n0-1 .. n0+16
  __shared__ float lout[(MODE == 0) ? 16 * H_ : 32];   // conv result for LN
  int b = blockIdx.y;
  int n0 = blockIdx.x * 16;
  const float* xb = x + (long long)b * N_ * H_;
  for (int i = threadIdx.x; i < 18 * H_; i += 256) {
    int rrow = i >> 8;
    int ch = i & 255;
    int p = n0 - 1 + rrow;
    lin[i] = (p >= 0 && p < N_) ? xb[(long long)p * H_ + ch] : 0.f;
  }
  __syncthreads();

  int wave = threadIdx.x >> 5;
  int lane = threadIdx.x & 31;
  int m = lane & 15;
  int kp = lane >> 4;  // K sub-pair select per WMMA f32 A/B layout
  int col0 = (wave * 2) * 16 + m;
  int col1 = (wave * 2 + 1) * 16 + m;
  const v2f* W2 = (const v2f*)Wp;
  v8f acc0 = {}, acc1 = {};
  for (int tap = 0; tap < 3; ++tap) {
    const float* arow = &lin[(m + tap) * H_];
    const v2f* wrow = W2 + (long long)tap * 128 * H_;
    if (tap < 2)  // keep next tap's weight block streaming into cache
      __builtin_prefetch((const void*)(wrow + 128 * H_ + col0), 0, 1);
    #pragma unroll 8
    for (int c = 0; c < H_; c += 4) {
      int k0 = c + 2 * kp;
      v2f a = *(const v2f*)(arow + k0);            // ds_load_b64
      v2f b0 = wrow[(k0 >> 1) * H_ + col0];        // global_load_b64, coalesced
      v2f b1 = wrow[(k0 >> 1) * H_ + col1];
      acc0 = WMMA_F32X4(a, b0, acc0);
      acc1 = WMMA_F32X4(a, b1, acc1);
    }
  }
  float bi0 = bias[col0], bi1 = bias[col1];

  if (MODE == 0) {
    #pragma unroll
    for (int r = 0; r < 8; ++r) {
      int row = r + 8 * kp;              // C/D layout: VGPR r, lanes16-31 -> M+8
      lout[row * H_ + col0] = acc0[r] + bi0;
      lout[row * H_ + col1] = acc1[r] + bi1;
    }
    __syncthreads();
    // LayerNorm + leaky-relu + residual; each wave owns 2 rows
    for (int rr = 0; rr < 2; ++rr) {
      int row = wave * 2 + rr;
      int p = n0 + row;
      const float* lr = &lout[row * H_];
      float s = 0.f, s2 = 0.f;
      #pragma unroll
      for (int i = lane; i < H_; i += 32) { float v = lr[i]; s += v; s2 = fmaf(v, v, s2); }
      #pragma unroll
      for (int off = 16; off > 0; off >>= 1) {
        s += __shfl_down(s, off, 32);
        s2 += __shfl_down(s2, off, 32);
      }
      s = __shfl(s, 0, 32);
      s2 = __shfl(s2, 0, 32);
      float mean = s * (1.f / H_);
      float var = s2 * (1.f / H_) - mean * mean;
      float rstd = rsqrtf(var + 1e-5f);
      if (p < N_) {
        float* op = out + ((long long)b * N_ + p) * H_;
        const float* xc = &lin[(row + 1) * H_];
        #pragma unroll
        for (int i = lane; i < H_; i += 32) {
          float hh = (lr[i] - mean) * rstd * lng[i] + lnb[i];
          hh = hh > 0.f ? hh : 0.01f * hh;
          op[i] = xc[i] + hh;
        }
      }
    }
  } else {
    #pragma unroll
    for (int r = 0; r < 8; ++r) {
      int row = r + 8 * kp;
      int p = n0 + row;
      if (p < N_) {
        float* op = out + ((long long)b * N_ + p) * H_;
        op[col0] = acc0[r] + bi0;
        op[col1] = acc1[r] + bi1;
      }
    }
  }
}

// ---------------------------------------------------------------------------
// 5) Pitch projection [.,256]x[256,104] via WMMA; weights pre-packed & padded
//    to 128 cols, so the K-loop is branch-free (EXEC stays all-ones).
// ---------------------------------------------------------------------------
__global__ __launch_bounds__(256) void k_pitch(
    const float* __restrict__ x, const float* __restrict__ wp_packed,
    const float* __restrict__ bp, float* __restrict__ logits,
    float* __restrict__ pred) {
  __shared__ float lin[16 * H_];
  __shared__ float llog[16 * NCP];
  int b = blockIdx.y;
  int n0 = blockIdx.x * 16;
  const float* xb = x + (long long)b * N_ * H_;
  for (int i = threadIdx.x; i < 16 * H_; i += 256) {
    int row = i >> 8;
    int ch = i & 255;
    int p = n0 + row;
    lin[i] = (p < N_) ? xb[(long long)p * H_ + ch] : 0.f;
  }
  __syncthreads();

  int wave = threadIdx.x >> 5;
  int lane = threadIdx.x & 31;
  int m = lane & 15;
  int kp = lane >> 4;
  int col = wave * 16 + m;                 // wave = column tile 0..7 (128 cols)
  const v2f* W2 = (const v2f*)wp_packed;   // [(k/2)*NCP + col]
  const float* arow = &lin[m * H_];
  v8f acc = {};
  #pragma unroll 8
  for (int c = 0; c < H_; c += 4) {
    int k0 = c + 2 * kp;
    v2f a = *(const v2f*)(arow + k0);      // ds_load_b64
    v2f bb = W2[(k0 >> 1) * NCP + col];    // global_load_b64, coalesced
    acc = WMMA_F32X4(a, bb, acc);
  }
  bool cv = (col < NCLS);
  float bc = cv ? bp[col] : 0.f;
  #pragma unroll
  for (int r = 0; r < 8; ++r) {
    int row = r + 8 * kp;
    float v = acc[r] + bc;
    llog[row * NCP + col] = cv ? v : -1e30f;
    int p = n0 + row;
    if (cv && p < N_)
      logits[((long long)b * N_ + p) * NCLS + col] = v;
  }
  __syncthreads();
  if (threadIdx.x < 16) {
    int row = threadIdx.x;
    int p = n0 + row;
    if (p < N_) {
      const float* lr = &llog[row * NCP];
      int bi = 0;
      float bv = lr[0];
      for (int j = 1; j < NCLS; ++j) {
        float v = lr[j];
        if (v > bv) { bv = v; bi = j; }   // first-occurrence tie-break
      }
      if (bi > 100 || bi < 30) bi = 0;    // NOTE_NUM=100, NOTE_START=30
      pred[(long long)b * N_ + p] = (float)bi;
    }
  }
}

// ---------------------------------------------------------------------------
extern "C" void kernel_launch(void* const* d_in, const int* in_sizes, int n_in,
                              void* d_out, int out_size, void* d_ws, size_t ws_size,
                              hipStream_t stream) {
  const float* feat        = (const float*)d_in[0];
  const int*   non_padding = (const int*)d_in[1];
  const float* w_bd        = (const float*)d_in[2];
  const float* b_bd        = (const float*)d_in[3];
  const float* w_attn      = (const float*)d_in[4];
  const float* b_attn      = (const float*)d_in[5];
  const float* conv_w      = (const float*)d_in[6];
  const float* conv_b      = (const float*)d_in[7];
  const float* ln_g        = (const float*)d_in[8];
  const float* ln_b        = (const float*)d_in[9];
  const float* post_w      = (const float*)d_in[10];
  const float* post_b      = (const float*)d_in[11];
  const float* w_pitch     = (const float*)d_in[12];
  const float* b_pitch     = (const float*)d_in[13];

  // d_out layout: concatenated reference return tuple, all as f32
  float* out = (float*)d_out;
  float* out_bd_logits = out;                                   // B*T
  float* out_bd        = out_bd_logits + (long long)B_ * T_;    // B*T
  float* out_len       = out_bd + (long long)B_ * T_;           // B
  float* out_logits    = out_len + B_;                          // B*N*104
  float* out_pred      = out_logits + (long long)B_ * N_ * NCLS;// B*N

  float* ws = (float*)d_ws;
  long long off = 0;
  float* probs     = ws + off; off += (long long)B_ * T_;
  float* attn_mean = ws + off; off += (long long)B_ * T_;
  int*   mel2note  = (int*)(ws + off); off += (long long)B_ * T_;
  int*   res       = (int*)(ws + off); off += (long long)B_ * T_;
  float* agg       = ws + off; off += (long long)B_ * N_ * H_;
  float* x2        = ws + off; off += (long long)B_ * N_ * H_;
  float* agg2      = ws + off; off += (long long)B_ * N_ * H_;
  float* denom     = ws + off; off += (long long)B_ * N_;
  float* wp_conv   = ws + off; off += (long long)3 * 128 * H_ * 2;  // packed conv_w
  float* wp_post   = ws + off; off += (long long)3 * 128 * H_ * 2;  // packed post_w
  float* wp_pitch  = ws + off; off += (long long)128 * NCP * 2;     // packed+padded w_pitch

  // 0) pack weights for WMMA-B fragment layout (runs before the GEMVs, same stream)
  {
    long long n = (long long)(768 / 2) * H_;   // conv K=768 (3 taps * 256)
    k_packw<<<(int)((n + 255) / 256), 256, 0, stream>>>(conv_w, wp_conv, 768, H_, H_);
    k_packw<<<(int)((n + 255) / 256), 256, 0, stream>>>(post_w, wp_post, 768, H_, H_);
    long long np = (long long)(H_ / 2) * NCP;
    k_packw<<<(int)((np + 255) / 256), 256, 0, stream>>>(w_pitch, wp_pitch, H_, NCLS, NCP);
  }
  // 1) boundary + attention GEMVs
  {
    long long rows = (long long)B_ * T_;
    int blocks = (int)((rows + 7) / 8);
    k_gemv<<<blocks, 256, 0, stream>>>(feat, w_bd, b_bd, w_attn, b_attn,
                                       out_bd_logits, probs, attn_mean);
  }
  // 2) zero accumulators
  {
    long long n = (long long)B_ * N_ * H_;
    k_zero<<<(int)((n + 255) / 256), 256, 0, stream>>>(agg, n);
    long long nd = (long long)B_ * N_;
    k_zero<<<(int)((nd + 255) / 256), 256, 0, stream>>>(denom, nd);
  }
  // 3) serial regulate_boundary + cumsum
  k_boundary<<<1, 32, 0, stream>>>(probs, non_padding, res, mel2note,
                                   out_bd, out_len);
  // 4) segment scatter
  {
    long long n = (long long)B_ * T_ * H_;
    k_scatter<<<(int)((n + 255) / 256), 256, 0, stream>>>(feat, attn_mean,
                                                          mel2note, agg, denom);
  }
  // 5) normalize
  {
    long long n = (long long)B_ * N_ * H_;
    k_div<<<(int)((n + 255) / 256), 256, 0, stream>>>(agg, denom);
  }
  // 6) residual conv block (conv + LN + leaky + residual)
  dim3 cg((N_ + 15) / 16, B_);
  k_conv<0><<<cg, 256, 0, stream>>>(agg, wp_conv, conv_b, ln_g, ln_b, x2);
  // 7) post conv
  k_conv<1><<<cg, 256, 0, stream>>>(x2, wp_post, post_b, nullptr, nullptr, agg2);
  // 8) pitch logits + argmax
  k_pitch<<<cg, 256, 0, stream>>>(agg2, wp_pitch, b_pitch, out_logits, out_pred);
}